// PartialChamferLoss_4492535791982
// MI455X (gfx1250) — compile-verified
//
#include <hip/hip_runtime.h>
#include <math.h>

// CDNA5 / gfx1250 Chamfer loss via V_WMMA_F32_16X16X4_F32.
// d(m,n) = |x_m|^2 + |y_n|^2 - 2 x_m.y_n ; the dot-product term is a K=3 GEMM
// done 16x16 per wave with one f32 WMMA (K padded to 4 with zeros).

typedef float v2f __attribute__((ext_vector_type(2)));
typedef float v8f __attribute__((ext_vector_type(8)));

#define BATCH 4
#define NPTS  8192          // points per cloud (N == M in the reference)
#define TILE  16
#define WPB   8             // waves per 256-thread block (wave32)

__global__ void chamfer_init_kernel(float* out) { out[0] = 0.0f; }

// One wave owns a 16-row strip of query points and sweeps all 8192 targets of
// its batch in 16-wide column tiles. Per tile: 1 point-load, 1 WMMA, 8 fma +
// 8 min. Running per-lane minima of (|y|^2 - 2 x.y) are min-reduced across the
// 16-lane halves at the end, |x|^2 is added back, sqrt'ed, and atomically
// accumulated (scaled) into the scalar loss.
__global__ __launch_bounds__(256) void chamfer_dir_kernel(
    const float* __restrict__ q,   // queries  [BATCH*NPTS, 3]
    const float* __restrict__ t,   // targets  [BATCH*NPTS, 3]
    float* __restrict__ out,       // scalar accumulator
    float scale)                   // 0.5 / (BATCH*NPTS)
{
  const int lane   = threadIdx.x & 31;
  const int wave   = threadIdx.x >> 5;
  const int waveId = blockIdx.x * WPB + wave;   // 0 .. 2047 (exact, no guard)
  const int row0   = waveId * TILE;             // first query row of this wave
  const int batch  = row0 / NPTS;
  const int tbase  = batch * NPTS;              // target rows of same cloud

  // A fragment: 16x4 f32. Lanes 0-15 carry K={0,1}=(x,y), lanes 16-31 K={2,3}=(z,0).
  const int  mrow = row0 + (lane & 15);
  const float ax = q[3 * mrow + 0];
  const float ay = q[3 * mrow + 1];
  const float az = q[3 * mrow + 2];
  v2f afrag;
  if (lane < 16) { afrag.x = ax; afrag.y = ay; }
  else           { afrag.x = az; afrag.y = 0.0f; }

  float m_min[8];
#pragma unroll
  for (int v = 0; v < 8; ++v) m_min[v] = 3.0e38f;

  for (int col0 = 0; col0 < NPTS; col0 += TILE) {
    // Each lane loads target point (col0 + lane%16); both halves need the
    // same point (different K slots), L0 makes the duplicate read free.
    const int   j  = tbase + col0 + (lane & 15);
    const float tx = t[3 * j + 0];
    const float ty = t[3 * j + 1];
    const float tz = t[3 * j + 2];
    const float y2 = tx * tx + ty * ty + tz * tz;

    v2f bfrag;  // B fragment: 4x16 f32, K split across lane halves like A.
    if (lane < 16) { bfrag.x = tx; bfrag.y = ty; }
    else           { bfrag.x = tz; bfrag.y = 0.0f; }

    v8f c = {};
    // (neg_a, A, neg_b, B, c_mod, C, reuse_a, reuse_b)
    c = __builtin_amdgcn_wmma_f32_16x16x4_f32(false, afrag, false, bfrag,
                                              (short)0, c, false, false);

    // Row m is fixed per (register, lane-half); col = lane%16. Track the
    // row-relevant part of the distance: |y|^2 - 2 * dot.
#pragma unroll
    for (int v = 0; v < 8; ++v) {
      const float d = __builtin_fmaf(c[v], -2.0f, y2);
      m_min[v] = fminf(m_min[v], d);
    }
  }

  // Min-reduce over the 16 lanes of each half (masks < 16 stay in-half).
#pragma unroll
  for (int v = 0; v < 8; ++v) {
#pragma unroll
    for (int m = 1; m < 16; m <<= 1)
      m_min[v] = fminf(m_min[v], __shfl_xor(m_min[v], m, 32));
  }

  // register v, half h -> query row = row0 + v + 8*h. One lane per half emits.
  float local = 0.0f;
  if ((lane & 15) == 0) {
    const int rbase = row0 + ((lane >> 4) << 3);
#pragma unroll
    for (int v = 0; v < 8; ++v) {
      const int   r  = rbase + v;
      const float qx = q[3 * r + 0];
      const float qy = q[3 * r + 1];
      const float qz = q[3 * r + 2];
      const float x2 = qx * qx + qy * qy + qz * qz;
      float d2 = m_min[v] + x2;
      d2 = d2 > 0.0f ? d2 : 0.0f;   // guard tiny negative from cancellation
      local += sqrtf(d2);
    }
  }
  local += __shfl_xor(local, 16, 32);   // combine the two halves
  if (lane == 0) atomicAdd(out, local * scale);
}

extern "C" void kernel_launch(void* const* d_in, const int* in_sizes, int n_in,
                              void* d_out, int out_size, void* d_ws, size_t ws_size,
                              hipStream_t stream) {
  (void)in_sizes; (void)n_in; (void)d_ws; (void)ws_size; (void)out_size;

  const float* xyz    = (const float*)d_in[0];   // [B*N, 3]
  const float* xyz_gt = (const float*)d_in[1];   // [B*M, 3]
  float* out = (float*)d_out;                    // scalar f32 loss

  const float scale = 0.5f / (float)(BATCH * NPTS);

  // 2048 waves per direction -> 256 blocks of 8 waves, exact coverage.
  const dim3 grid(BATCH * NPTS / (TILE * WPB));  // 256
  const dim3 block(32 * WPB);                    // 256 threads

  chamfer_init_kernel<<<1, 1, 0, stream>>>(out);
  // direction: for each predicted point, nearest gt point  (loss_)
  chamfer_dir_kernel<<<grid, block, 0, stream>>>(xyz, xyz_gt, out, scale);
  // direction: for each gt point, nearest predicted point  (loss)
  chamfer_dir_kernel<<<grid, block, 0, stream>>>(xyz_gt, xyz, out, scale);
}